// RealAgnosticResidualInteractionBlock_89979564851569
// MI455X (gfx1250) — compile-verified
//
#include <hip/hip_runtime.h>
#include <math.h>
#include <stdint.h>

typedef float v2f __attribute__((ext_vector_type(2)));
typedef float v8f __attribute__((ext_vector_type(8)));

#define N_NODES 8000
#define N_EDGES 200000

// CDNA5 async global->LDS copies (ASYNCcnt) if this toolchain exposes them.
#if defined(__AMDGCN__) && __has_builtin(__builtin_amdgcn_global_load_async_to_lds_b128) && \
    __has_builtin(__builtin_amdgcn_s_wait_asynccnt)
#define HAVE_ASYNC 1
#else
#define HAVE_ASYNC 0
#endif

#if HAVE_ASYNC
typedef int v4i __attribute__((vector_size(16)));
typedef __attribute__((address_space(1))) v4i gv4i;  // global (cuda_device)
typedef __attribute__((address_space(3))) v4i lv4i;  // LDS
#endif

__device__ __forceinline__ void stage16(float* dst_lds, const float* src) {
#if HAVE_ASYNC
  // Global VA is identity in AS1; generic LDS pointer carries the LDS byte
  // offset in its low 32 bits (ISA 10.2), and AS3 pointers are 32-bit.
  __builtin_amdgcn_global_load_async_to_lds_b128(
      (gv4i*)(uintptr_t)(const void*)src,
      (lv4i*)(uint32_t)(uintptr_t)(void*)dst_lds, 0, 0);
#else
  *(float4*)dst_lds = *(const float4*)src;
#endif
}

__device__ __forceinline__ void wait_async0() {
#if HAVE_ASYNC
  __builtin_amdgcn_s_wait_asynccnt(0);
#endif
}

__device__ __forceinline__ v8f vzero8() {
  v8f z = {0.f, 0.f, 0.f, 0.f, 0.f, 0.f, 0.f, 0.f};
  return z;
}

__device__ __forceinline__ float silu_c(float x) {
  return (x / (1.0f + __expf(-x))) * 1.6790512f;  // silu(x) * SILU_C
}

// One wave computes a 16x16 f32 tile via V_WMMA_F32_16X16X4_F32.
// A: row-major, 16 rows, row stride lda, element stride as.
// B: row-major KxN slab, pre-offset to column n0, row stride ldb.
__device__ __forceinline__ v8f gemm16(const float* A, int lda, int as,
                                      const float* B, int ldb, int K, v8f acc) {
  const int lane = threadIdx.x & 31;
  const int lm = lane & 15;
  const int lh = lane >> 4;
  const float* ar = A + lm * lda;
  const float* bc = B + lm;
  for (int kk = 0; kk < K; kk += 4) {
    const int k0 = kk + 2 * lh;
    v2f a, b;
    a.x = ar[k0 * as];
    a.y = ar[(k0 + 1) * as];
    b.x = bc[k0 * ldb];
    b.y = bc[(k0 + 1) * ldb];
    acc = __builtin_amdgcn_wmma_f32_16x16x4_f32(false, a, false, b, (short)0, acc,
                                                false, false);
  }
  return acc;
}

__device__ __forceinline__ void store16(float* D, int rs, int cs, v8f d, float scale) {
  const int lane = threadIdx.x & 31;
  const int lm = lane & 15;
  const int lh = lane >> 4;
#pragma unroll
  for (int r = 0; r < 8; ++r)
    D[(r + 8 * lh) * rs + lm * cs] = d[r] * scale;
}

// ---------------------------------------------------------------------------
__global__ void k_zero(float* p, size_t n) {
  size_t i = (size_t)blockIdx.x * blockDim.x + threadIdx.x;
  const size_t stride = (size_t)gridDim.x * blockDim.x;
  for (; i < n; i += stride) p[i] = 0.0f;
}

// s_up = s @ w_up_0e / sqrt(128); v_up[n,w,m] = sum_u v[n,u,m] w_up_1o[u,w] / sqrt(128)
__global__ void __launch_bounds__(256) k_up(const float* __restrict__ nf,
                                            const float* __restrict__ w0e,
                                            const float* __restrict__ w1o,
                                            float* __restrict__ s_up,
                                            float* __restrict__ v_up) {
  const int m0 = blockIdx.x * 16;
  const int g = blockIdx.y;  // 0: scalar, 1..3: vector comp m
  const int n0 = (threadIdx.x >> 5) * 16;
  v8f acc = vzero8();
  const float sc = 0.08838834764831845f;  // 1/sqrt(128)
  if (g == 0) {
    acc = gemm16(nf + m0 * 512, 512, 1, w0e + n0, 128, 128, acc);
    store16(s_up + m0 * 128 + n0, 128, 1, acc, sc);
  } else {
    const int m = g - 1;
    acc = gemm16(nf + m0 * 512 + 128 + m, 512, 3, w1o + n0, 128, 128, acc);
    store16(v_up + m0 * 384 + n0 * 3 + m, 384, 3, acc, sc);
  }
}

// Skip connection: sc = [(s (x) attrs) @ Wskip0e, per-m (v_m (x) attrs) @ Wskip1o]
// Outer-product rows (K=1280) built chunk-wise (K-chunk 160 = 16 u * 10 v) in LDS.
__global__ void __launch_bounds__(256) k_skip(const float* __restrict__ nf,
                                              const float* __restrict__ attrs,
                                              const float* __restrict__ wsk0,
                                              const float* __restrict__ wsk1,
                                              float* __restrict__ sc) {
  __shared__ float At[16][164];  // padded: bank-conflict-free b64 reads
  const int m0 = blockIdx.x * 16;
  const int g = blockIdx.y;
  const float* W = (g == 0) ? wsk0 : wsk1;  // (1280,128): row u*10+v, col w
  const int n0 = (threadIdx.x >> 5) * 16;
  v8f acc = vzero8();
  for (int u0 = 0; u0 < 128; u0 += 16) {
    __syncthreads();
    for (int idx = threadIdx.x; idx < 16 * 160; idx += 256) {
      const int row = idx / 160;
      const int kc = idx - row * 160;  // uu*10 + v
      const int uu = kc / 10;
      const int v = kc - uu * 10;
      const int node = m0 + row;
      const float f = (g == 0) ? nf[node * 512 + u0 + uu]
                               : nf[node * 512 + 128 + (u0 + uu) * 3 + (g - 1)];
      At[row][kc] = f * attrs[node * 10 + v];
    }
    __syncthreads();
    acc = gemm16(&At[0][0], 164, 1, W + u0 * 10 * 128 + n0, 128, 160, acc);
  }
  const float cs = 0.027950849718747373f;  // 1/sqrt(1280)
  if (g == 0)
    store16(sc + m0 * 512 + n0, 512, 1, acc, cs);
  else
    store16(sc + m0 * 512 + 128 + n0 * 3 + (g - 1), 512, 3, acc, cs);
}

// Fused: edge MLP (WMMA, LDS-staged weights with async double-buffering)
// -> per-edge messages -> scatter-add into L2-resident agg buffers.
// 4 waves/block, each wave owns 16 edges end-to-end; tpw never touches HBM.
__global__ void __launch_bounds__(128) k_edge(const float* __restrict__ eattr,
                                              const float* __restrict__ efeat,
                                              const int* __restrict__ eidx,
                                              const float* __restrict__ wr1,
                                              const float* __restrict__ wr2,
                                              const float* __restrict__ wr3,
                                              const float* __restrict__ wr4,
                                              const float* __restrict__ s_up,
                                              const float* __restrict__ v_up,
                                              float* __restrict__ agg0,
                                              float* __restrict__ agg1) {
  __shared__ float hA[4][16][68];   // per-wave activations
  __shared__ float hB[4][16][68];
  __shared__ float Wbuf[64][68];    // staged w_r2 / w_r3 (padded rows, 16B aligned)
  __shared__ float W4[2][64][20];   // double-buffered w_r4 column slab (64x16)
  __shared__ float y0s[4][16];
  __shared__ float y1s[4][16][3];
  __shared__ int snds[4][16];
  __shared__ int rcvs[4][16];

  const int wave = threadIdx.x >> 5;
  const int lane = threadIdx.x & 31;
  const int lm = lane & 15;
  const int lh = lane >> 4;
  const int e0 = (blockIdx.x * 4 + wave) * 16;

  if (lane < 16) {  // per-wave metadata; same-wave LDS ops are in-order
    const int e = e0 + lane;
    y0s[wave][lane] = eattr[e * 4 + 0];
    y1s[wave][lane][0] = eattr[e * 4 + 1];
    y1s[wave][lane][1] = eattr[e * 4 + 2];
    y1s[wave][lane][2] = eattr[e * 4 + 3];
    snds[wave][lane] = eidx[e];
    rcvs[wave][lane] = eidx[N_EDGES + e];
  }

  // Stage w_r2 (64x64) asynchronously; overlap with layer 1 compute.
  for (int i = threadIdx.x; i < 1024; i += 128)
    stage16(&Wbuf[i >> 4][(i & 15) * 4], wr2 + (i >> 4) * 64 + (i & 15) * 4);

  // Layer 1: (16x8) @ (8x64) from global wr1, scale 1/sqrt(8), silu -> hA
  for (int nt = 0; nt < 4; ++nt) {
    v8f acc = vzero8();
    acc = gemm16(efeat + e0 * 8, 8, 1, wr1 + nt * 16, 64, 8, acc);
#pragma unroll
    for (int r = 0; r < 8; ++r)
      hA[wave][r + 8 * lh][nt * 16 + lm] = silu_c(acc[r] * 0.3535533905932738f);
  }
  wait_async0();
  __syncthreads();  // Wbuf ready for all waves

  // Layer 2: hA @ Wbuf(w_r2) -> hB
  for (int nt = 0; nt < 4; ++nt) {
    v8f acc = vzero8();
    acc = gemm16(&hA[wave][0][0], 68, 1, &Wbuf[0][nt * 16], 68, 64, acc);
#pragma unroll
    for (int r = 0; r < 8; ++r)
      hB[wave][r + 8 * lh][nt * 16 + lm] = silu_c(acc[r] * 0.125f);
  }
  __syncthreads();  // all waves done reading w_r2 copy

  // Stage w_r3 into Wbuf and the first w_r4 slab into W4[0].
  for (int i = threadIdx.x; i < 1024; i += 128)
    stage16(&Wbuf[i >> 4][(i & 15) * 4], wr3 + (i >> 4) * 64 + (i & 15) * 4);
  for (int i = threadIdx.x; i < 256; i += 128)
    stage16(&W4[0][i >> 2][(i & 3) * 4], wr4 + (i >> 2) * 512 + (i & 3) * 4);
  wait_async0();
  __syncthreads();

  // Layer 3: hB @ Wbuf(w_r3) -> hA
  for (int nt = 0; nt < 4; ++nt) {
    v8f acc = vzero8();
    acc = gemm16(&hB[wave][0][0], 68, 1, &Wbuf[0][nt * 16], 68, 64, acc);
#pragma unroll
    for (int r = 0; r < 8; ++r)
      hA[wave][r + 8 * lh][nt * 16 + lm] = silu_c(acc[r] * 0.125f);
  }

  // Layer 4 fused with messages; double-buffered async slab prefetch.
  // tile nt -> chunk c = nt/8 selects w0/w1/w2/w3; u0 = (nt%8)*16 channels.
  for (int nt = 0; nt < 32; ++nt) {
    const int buf = nt & 1;
    if (nt + 1 < 32) {  // prefetch next slab into the other buffer
      const int c1 = (nt + 1) >> 3;
      const int u1 = ((nt + 1) & 7) * 16;
      const float* src = wr4 + c1 * 128 + u1;
      for (int i = threadIdx.x; i < 256; i += 128)
        stage16(&W4[buf ^ 1][i >> 2][(i & 3) * 4], src + (i >> 2) * 512 + (i & 3) * 4);
    }
    const int c = nt >> 3;
    const int u0 = (nt & 7) * 16;
    v8f acc = vzero8();
    acc = gemm16(&hA[wave][0][0], 68, 1, &W4[buf][0][0], 20, 64, acc);
    const int u = u0 + lm;
#pragma unroll
    for (int r = 0; r < 8; ++r) {
      const int ei = r + 8 * lh;        // edge within the wave's 16
      const float w = acc[r] * 0.125f;  // tpw element
      const int snd = snds[wave][ei];
      const int rcv = rcvs[wave][ei];
      const float y0 = y0s[wave][ei];
      if (c == 0) {  // m0a = w0 * xs * y0 -> agg0[:, u]
        const float xs = s_up[snd * 128 + u];
        atomicAdd(&agg0[rcv * 256 + u], w * xs * y0);
      } else if (c == 1) {  // m0b = w1 * (xv . y1)/sqrt(3) -> agg0[:, 128+u]
        const float* xv = &v_up[(snd * 128 + u) * 3];
        const float d = xv[0] * y1s[wave][ei][0] + xv[1] * y1s[wave][ei][1] +
                        xv[2] * y1s[wave][ei][2];
        atomicAdd(&agg0[rcv * 256 + 128 + u], w * d * 0.5773502691896258f);
      } else if (c == 2) {  // m1a = w2 * xs * y1[m] -> agg1[:, u, m]
        const float t = w * s_up[snd * 128 + u];
        atomicAdd(&agg1[(rcv * 256 + u) * 3 + 0], t * y1s[wave][ei][0]);
        atomicAdd(&agg1[(rcv * 256 + u) * 3 + 1], t * y1s[wave][ei][1]);
        atomicAdd(&agg1[(rcv * 256 + u) * 3 + 2], t * y1s[wave][ei][2]);
      } else {  // m1b = w3 * xv[m] * y0 -> agg1[:, 128+u, m]
        const float* xv = &v_up[(snd * 128 + u) * 3];
        const float t = w * y0;
        atomicAdd(&agg1[(rcv * 256 + 128 + u) * 3 + 0], t * xv[0]);
        atomicAdd(&agg1[(rcv * 256 + 128 + u) * 3 + 1], t * xv[1]);
        atomicAdd(&agg1[(rcv * 256 + 128 + u) * 3 + 2], t * xv[2]);
      }
    }
    wait_async0();    // next slab landed
    __syncthreads();  // all waves done with W4[buf] before it is restaged
  }
}

// Output GEMMs: out0 = agg0 @ w_out_0e; out1_m = agg1[:,:,m] @ w_out_1o.
__global__ void __launch_bounds__(256) k_out(const float* __restrict__ agg0,
                                             const float* __restrict__ agg1,
                                             const float* __restrict__ w0e,
                                             const float* __restrict__ w1o,
                                             float* __restrict__ msg) {
  const int m0 = blockIdx.x * 16;
  const int g = blockIdx.y;
  const int n0 = (threadIdx.x >> 5) * 16;
  v8f acc = vzero8();
  const float co = 0.0025f;  // 1/(sqrt(256)*25)
  if (g == 0) {
    acc = gemm16(agg0 + m0 * 256, 256, 1, w0e + n0, 128, 256, acc);
    store16(msg + m0 * 512 + n0 * 4, 512, 4, acc, co);
  } else {
    const int m = g - 1;
    acc = gemm16(agg1 + m0 * 768 + m, 768, 3, w1o + n0, 128, 256, acc);
    store16(msg + m0 * 512 + n0 * 4 + g, 512, 4, acc, co);
  }
}

// ---------------------------------------------------------------------------
extern "C" void kernel_launch(void* const* d_in, const int* in_sizes, int n_in,
                              void* d_out, int out_size, void* d_ws, size_t ws_size,
                              hipStream_t stream) {
  const float* node_attrs = (const float*)d_in[0];
  const float* node_feats = (const float*)d_in[1];
  const float* edge_attrs = (const float*)d_in[2];
  const float* edge_feats = (const float*)d_in[3];
  const int* edge_index = (const int*)d_in[4];
  const float* w_up_0e = (const float*)d_in[5];
  const float* w_up_1o = (const float*)d_in[6];
  const float* w_skip_0e = (const float*)d_in[7];
  const float* w_skip_1o = (const float*)d_in[8];
  const float* w_r1 = (const float*)d_in[9];
  const float* w_r2 = (const float*)d_in[10];
  const float* w_r3 = (const float*)d_in[11];
  const float* w_r4 = (const float*)d_in[12];
  const float* w_out_0e = (const float*)d_in[13];
  const float* w_out_1o = (const float*)d_in[14];

  float* msg = (float*)d_out;               // 8000*128*4
  float* sc = msg + (size_t)N_NODES * 512;  // 8000*512

  float* ws = (float*)d_ws;
  float* s_up = ws;                            // 8000*128
  float* v_up = s_up + (size_t)N_NODES * 128;  // 8000*384  (n,u,m)
  float* agg0 = v_up + (size_t)N_NODES * 384;  // 8000*256
  float* agg1 = agg0 + (size_t)N_NODES * 256;  // 8000*768  (n,c,m)

  k_zero<<<2048, 256, 0, stream>>>(agg0, (size_t)N_NODES * 1024);
  k_up<<<dim3(500, 4), 256, 0, stream>>>(node_feats, w_up_0e, w_up_1o, s_up, v_up);
  k_skip<<<dim3(500, 4), 256, 0, stream>>>(node_feats, node_attrs, w_skip_0e,
                                           w_skip_1o, sc);
  k_edge<<<3125, 128, 0, stream>>>(edge_attrs, edge_feats, edge_index, w_r1, w_r2,
                                   w_r3, w_r4, s_up, v_up, agg0, agg1);
  k_out<<<dim3(500, 4), 256, 0, stream>>>(agg0, agg1, w_out_0e, w_out_1o, msg);
}